// LinkPredictor_72232759984607
// MI455X (gfx1250) — compile-verified
//
#include <hip/hip_runtime.h>

#define F 128

typedef __attribute__((ext_vector_type(2))) float v2f;
typedef __attribute__((ext_vector_type(8))) float v8f;

// ---------------- zero fill (float4 granularity) ----------------
__global__ void zero_f4_kernel(float4* __restrict__ p, int n4) {
    int i = blockIdx.x * blockDim.x + threadIdx.x;
    if (i < n4) p[i] = make_float4(0.f, 0.f, 0.f, 0.f);
}

// ---------------- in-degree (as float) ----------------
__global__ void degree_kernel(const int* __restrict__ dst, float* __restrict__ deg, int nedges) {
    int i = blockIdx.x * blockDim.x + threadIdx.x;
    if (i < nedges) atomicAdd(&deg[dst[i]], 1.0f);
}

// ---------------- edge scatter-add: one wave32 per edge ----------------
// lane l moves bytes [16l, 16l+16) of the 512B feature row: float4 load + 4 f32 atomics.
__global__ void scatter_add_kernel(const float* __restrict__ X,
                                   const int* __restrict__ src,
                                   const int* __restrict__ dst,
                                   float* __restrict__ AGG, int nedges) {
    int gid  = blockIdx.x * blockDim.x + threadIdx.x;
    int e    = gid >> 5;
    int lane = gid & 31;
    if (e >= nedges) return;
    int s = src[e];
    int d = dst[e];
    const float4 v = *((const float4*)(X + (size_t)s * F) + lane);
    float* ag = AGG + (size_t)d * F + (size_t)lane * 4;
    atomicAdd(ag + 0, v.x);
    atomicAdd(ag + 1, v.y);
    atomicAdd(ag + 2, v.z);
    atomicAdd(ag + 3, v.w);
}

// ---------------- fused SAGE layer GEMM ----------------
// OUT[16 rows x 128 cols per wave] = act( X@Wself + (AGG/max(deg,1))@Wneigh + b )
// f32 WMMA 16x16x4: A 16x4 (v2f/lane), B 4x16 (v2f/lane), C/D 16x16 (v8f/lane).
__global__ void __launch_bounds__(256) sage_gemm_kernel(
    const float* __restrict__ X, const float* __restrict__ AGG,
    const float* __restrict__ DEG,
    const float* __restrict__ Wself, const float* __restrict__ Wneigh,
    const float* __restrict__ bias, float* __restrict__ OUT,
    int ntiles, int relu)
{
    const int wave = threadIdx.x >> 5;
    const int lane = threadIdx.x & 31;
    const int tile = blockIdx.x * 8 + wave;
    if (tile >= ntiles) return;               // uniform per wave -> EXEC stays all-ones

    const int row0   = tile * 16;
    const int laneLo = lane & 15;             // M index (A) / N index (B, C/D)
    const int laneHi = lane >> 4;             // K half-select for A/B fragments
    const int rA     = row0 + laneLo;         // A-matrix row owned by this lane
    const int kOff   = laneHi * 2;            // A/B hold K = k4+kOff, k4+kOff+1

    const float* xrow = X   + (size_t)rA * F + kOff;
    const float* arow = AGG + (size_t)rA * F + kOff;
    const float  inv  = 1.0f / fmaxf(DEG[rA], 1.0f);   // mean normalization, folded into A

    for (int nt = 0; nt < 8; ++nt) {
        const int n0  = nt * 16;
        const int col = n0 + laneLo;
        v8f acc = {};
        for (int k4 = 0; k4 < F; k4 += 4) {
            // self term
            v2f a = *(const v2f*)(xrow + k4);
            v2f bs;
            bs.x = Wself[(k4 + kOff)     * F + col];
            bs.y = Wself[(k4 + kOff + 1) * F + col];
            acc = __builtin_amdgcn_wmma_f32_16x16x4_f32(
                      false, a, false, bs, (short)0, acc, false, false);
            // neighbor term (degree-normalized)
            v2f an = *(const v2f*)(arow + k4);
            an.x *= inv; an.y *= inv;
            v2f bn;
            bn.x = Wneigh[(k4 + kOff)     * F + col];
            bn.y = Wneigh[(k4 + kOff + 1) * F + col];
            acc = __builtin_amdgcn_wmma_f32_16x16x4_f32(
                      false, an, false, bn, (short)0, acc, false, false);
        }
        const float bv = bias[col];
        #pragma unroll
        for (int r = 0; r < 8; ++r) {         // VGPR r -> row M = r (lanes 0-15) / r+8 (lanes 16-31)
            float v = acc[r] + bv;
            if (relu) v = fmaxf(v, 0.0f);
            OUT[(size_t)(row0 + laneHi * 8 + r) * F + col] = v;
        }
    }
}

extern "C" void kernel_launch(void* const* d_in, const int* in_sizes, int n_in,
                              void* d_out, int out_size, void* d_ws, size_t ws_size,
                              hipStream_t stream) {
    const float* x   = (const float*)d_in[0];
    const float* Ws1 = (const float*)d_in[1];
    const float* Wn1 = (const float*)d_in[2];
    const float* b1  = (const float*)d_in[3];
    const float* Ws2 = (const float*)d_in[4];
    const float* Wn2 = (const float*)d_in[5];
    const float* b2  = (const float*)d_in[6];
    const int*   src = (const int*)d_in[7];
    const int*   dst = (const int*)d_in[8];

    const int N = in_sizes[0] / F;   // 100000
    const int E = in_sizes[7];       // 1600000

    // workspace: deg[N] | agg[N*F] | h1[N*F]
    float* deg = (float*)d_ws;
    float* agg = deg + (size_t)((N + 3) & ~3);
    float* h1  = agg + (size_t)N * F;
    float* out = (float*)d_out;

    const int T = 256;
    const int nDeg4   = (N + 3) / 4;
    const int nAgg4   = N * (F / 4);
    const int ntiles  = N / 16;                       // 100000 % 16 == 0
    const int gemmBlk = (ntiles + 7) / 8;
    const int scatBlk = (E * 32 + T - 1) / T;         // one wave per edge

    // ---- degree (computed once, reused by both layers) ----
    zero_f4_kernel<<<(nDeg4 + T - 1) / T, T, 0, stream>>>((float4*)deg, nDeg4);
    degree_kernel<<<(E + T - 1) / T, T, 0, stream>>>(dst, deg, E);

    // ---- layer 1 ----
    zero_f4_kernel<<<(nAgg4 + T - 1) / T, T, 0, stream>>>((float4*)agg, nAgg4);
    scatter_add_kernel<<<scatBlk, T, 0, stream>>>(x, src, dst, agg, E);
    sage_gemm_kernel<<<gemmBlk, T, 0, stream>>>(x, agg, deg, Ws1, Wn1, b1, h1, ntiles, 1);

    // ---- layer 2 ----
    zero_f4_kernel<<<(nAgg4 + T - 1) / T, T, 0, stream>>>((float4*)agg, nAgg4);
    scatter_add_kernel<<<scatBlk, T, 0, stream>>>(h1, src, dst, agg, E);
    sage_gemm_kernel<<<gemmBlk, T, 0, stream>>>(h1, agg, deg, Ws2, Wn2, b2, out, ntiles, 0);
}